// GlobalPooling_2843268350276
// MI455X (gfx1250) — compile-verified
//
#include <hip/hip_runtime.h>
#include <hip/hip_bf16.h>
#include <math.h>

// Segment mean+max pooling, MI455X (gfx1250, wave32).
// x: [N,128] f32; batch: [N] sorted segment ids in [0,B); out: [B,256] f32.
// HBM-bound (~1.02 GB read -> ~44us floor at 23.3 TB/s). One workgroup per
// segment, binary search for the contiguous row range (deterministic, no
// atomics). Sum reduction runs on the matrix pipe via V_WMMA_F32_16X16X4_F32
// with an all-ones A matrix (full f32 precision); max pooling on VALU from
// the same loaded registers.

typedef __attribute__((ext_vector_type(2))) float v2f;
typedef __attribute__((ext_vector_type(8))) float v8f;

#define FDIM 128
#define TILES 8 // 128 features / 16 cols per WMMA tile

__device__ __forceinline__ int lower_bound_i32(const int* __restrict__ b,
                                               int n, int key) {
    int lo = 0, hi = n;
    while (lo < hi) {
        int mid = (lo + hi) >> 1;
        if (b[mid] < key) lo = mid + 1; else hi = mid;
    }
    return lo;
}

__global__ __launch_bounds__(128)
void seg_pool_wmma_kernel(const float* __restrict__ x,
                          const int* __restrict__ batch,
                          float* __restrict__ out,
                          int nrows) {
    const int seg  = blockIdx.x;
    const int tid  = threadIdx.x;
    const int wave = tid >> 5;   // 0..3
    const int lane = tid & 31;

    // Sorted batch -> contiguous [start, end) row range; count = end-start.
    const int start = lower_bound_i32(batch, nrows, seg);
    const int end   = lower_bound_i32(batch, nrows, seg + 1);

    // WMMA B-matrix (4x16 f32) lane mapping:
    //   lanes 0-15 hold rows {r, r+1}; lanes 16-31 hold rows {r+2, r+3};
    //   column n = lane & 15. A = ones => D[m][n] = sum_k B[k][n]
    //   (K-slot permutation irrelevant: we only use column sums).
    const int rowOff  = (lane >> 4) << 1; // 0 or 2
    const int colBase = lane & 15;

    const v8f zero8 = {};
    v8f   c[TILES];
    float mx[TILES];
#pragma unroll
    for (int t = 0; t < TILES; ++t) { c[t] = zero8; mx[t] = -INFINITY; }

    v2f ones; ones[0] = 1.0f; ones[1] = 1.0f;

    int r = start + 4 * wave;
    // Full 4-row chunks for this wave (wave-uniform trip count, EXEC all-1s).
    for (; r + 3 < end; r += 16) {
        const float* p0 = x + (size_t)(r + rowOff) * FDIM + colBase;
        const float* p1 = p0 + FDIM;
        const int rn = r + 16;
        if (rn + 3 < end) {
            // 64B per lane covers this wave's next 2KB chunk.
            __builtin_prefetch(x + (size_t)rn * FDIM + lane * 16, 0, 0);
        }
#pragma unroll
        for (int t = 0; t < TILES; ++t) {
            float b0 = p0[t * 16];
            float b1 = p1[t * 16];
            v2f bb; bb[0] = b0; bb[1] = b1;
            // (neg_a, A, neg_b, B, c_mod, C, reuse_a, reuse_b)
            c[t] = __builtin_amdgcn_wmma_f32_16x16x4_f32(
                false, ones, false, bb, (short)0, c[t], false, false);
            mx[t] = fmaxf(mx[t], fmaxf(b0, b1));
        }
    }
    // Predicated tail chunk (condition is wave-uniform; lane predicates are
    // selects, so EXEC stays all-1s for the WMMA).
    if (r < end) {
        const int  row0 = r + rowOff;
        const int  row1 = row0 + 1;
        const bool v0   = row0 < end;
        const bool v1   = row1 < end;
        const float* p0 = x + (size_t)row0 * FDIM + colBase;
        const float* p1 = p0 + FDIM;
#pragma unroll
        for (int t = 0; t < TILES; ++t) {
            float b0 = v0 ? p0[t * 16] : 0.0f;   // 0 for the sum path
            float b1 = v1 ? p1[t * 16] : 0.0f;
            v2f bb; bb[0] = b0; bb[1] = b1;
            c[t] = __builtin_amdgcn_wmma_f32_16x16x4_f32(
                false, ones, false, bb, (short)0, c[t], false, false);
            float m0 = v0 ? b0 : -INFINITY;      // -inf for the max path
            float m1 = v1 ? b1 : -INFINITY;
            mx[t] = fmaxf(mx[t], fmaxf(m0, m1));
        }
    }

    // Cross-wave combine in LDS. D row M=0 (feature sums) lives in c[t][0] of
    // lanes 0-15 (lanes 16-31 hold the identical M=8 row). Fold the half-wave
    // row groups of the max with a lane-16 xor shuffle.
    __shared__ float s_sum[4][FDIM];
    __shared__ float s_max[4][FDIM];
#pragma unroll
    for (int t = 0; t < TILES; ++t) {
        float mc = fmaxf(mx[t], __shfl_xor(mx[t], 16));
        if (lane < 16) {
            s_sum[wave][t * 16 + lane] = c[t][0];
            s_max[wave][t * 16 + lane] = mc;
        }
    }
    __syncthreads();

    const int f = tid; // 0..127, one feature per thread
    float s = (s_sum[0][f] + s_sum[1][f]) + (s_sum[2][f] + s_sum[3][f]);
    float m = fmaxf(fmaxf(s_max[0][f], s_max[1][f]),
                    fmaxf(s_max[2][f], s_max[3][f]));
    const int   cnt   = end - start;
    const float denom = (float)(cnt > 0 ? cnt : 1);
    const size_t o = (size_t)seg * (2 * FDIM);
    out[o + f]        = s / denom;  // mean (empty segment -> 0/1 = 0)
    out[o + FDIM + f] = m;          // max  (empty segment -> -inf)
}

extern "C" void kernel_launch(void* const* d_in, const int* in_sizes, int n_in,
                              void* d_out, int out_size, void* d_ws, size_t ws_size,
                              hipStream_t stream) {
    (void)n_in; (void)d_ws; (void)ws_size;
    const float* x     = (const float*)d_in[0];
    const int*   batch = (const int*)d_in[1];
    float*       out   = (float*)d_out;

    const int nrows = in_sizes[0] / FDIM;    // N
    const int nseg  = out_size / (2 * FDIM); // B

    dim3 grid(nseg), block(128);
    hipLaunchKernelGGL(seg_pool_wmma_kernel, grid, block, 0, stream,
                       x, batch, out, nrows);
}